// VoxelGrid_81320910782594
// MI455X (gfx1250) — compile-verified
//
#include <hip/hip_runtime.h>

// Voxel-grid scatter-mean for MI455X (gfx1250, wave32).
// Bandwidth-bound (~0.8-1.0 GB traffic -> ~35-45us @ 23.3 TB/s). No matmul
// content, so no WMMA; CDNA5-specific paths used instead:
//   * global_store_async_from_lds_b128 + s_wait_asynccnt (zeroing pass)
//   * no-return global_atomic_add_f32 at agent scope (scatter)
//   * LDS-staged, fully coalesced b128 non-temporal stores for the 320MB
//     streamed output (largest traffic term)

#define VSZ     100
#define NPOINTS (8 * 65536)                 // 524288 points
#define NVOX    (8 * VSZ * VSZ * VSZ)       // 8,000,000 voxels (divisible by 256)

typedef float v4f __attribute__((ext_vector_type(4)));

// ---------------------------------------------------------------------------
// Kernel 1: zero the accumulator using async LDS->global stores (ASYNCcnt).
// Each lane repeatedly issues a 16B async store sourced from a zeroed 512B
// LDS block. Low-32 bits of a flat shared-pointer are the raw LDS byte
// address (ISA 10.2).
// ---------------------------------------------------------------------------
__global__ void vg_zero_async(float* __restrict__ p, long long n4) {
  __shared__ float zbuf[128];               // 512 bytes of zeros
  const int t = threadIdx.x;
  if (t < 128) zbuf[t] = 0.0f;
  __syncthreads();

  const unsigned lds_off = (unsigned)(size_t)(&zbuf[(t & 31) * 4]); // 16B/lane

  long long i = (long long)blockIdx.x * blockDim.x + t;
  const long long stride = (long long)gridDim.x * blockDim.x;
  for (; i < n4; i += stride) {
    unsigned long long ga =
        (unsigned long long)(size_t)p + (unsigned long long)i * 16ull;
    asm volatile("global_store_async_from_lds_b128 %0, %1, off"
                 :
                 : "v"(ga), "v"(lds_off)
                 : "memory");
  }
  asm volatile("s_wait_asynccnt 0x0" ::: "memory");
}

// ---------------------------------------------------------------------------
// Voxel index: exact replica of the reference f32 math.
//   res = f32(1/100); fl = floor((c + res) / res); clip to [0, 101].
// ---------------------------------------------------------------------------
__device__ __forceinline__ int vg_voxel_idx(float c) {
  const float res = 0.01f;                  // == f32(1.0/100.0)
  float q = (c + res) / res;                // plain f32 add + IEEE f32 div
  int fl = (int)floorf(q);
  fl = fl < 0 ? 0 : fl;
  fl = fl > 101 ? 101 : fl;
  return fl;
}

// ---------------------------------------------------------------------------
// Kernel 2: one thread per point; 7 no-return fp32 atomics per point.
// Boundary cells (index 0 or 101) are sliced off by the reference -> skip.
// ---------------------------------------------------------------------------
__global__ void vg_scatter(const float* __restrict__ coords,
                           const float* __restrict__ feats,
                           float* __restrict__ acc, int astride) {
  const int p = blockIdx.x * blockDim.x + threadIdx.x;
  if (p >= NPOINTS) return;

  const long long o = (long long)p * 3;
  const float cx = coords[o + 0], cy = coords[o + 1], cz = coords[o + 2];
  const float f0 = feats[o + 0],  f1 = feats[o + 1],  f2 = feats[o + 2];

  const int vx = vg_voxel_idx(cx);
  const int vy = vg_voxel_idx(cy);
  const int vz = vg_voxel_idx(cz);
  if (vx < 1 || vx > VSZ || vy < 1 || vy > VSZ || vz < 1 || vz > VSZ) return;

  const int b = p >> 16;                    // NUM_COORDS == 65536
  const long long base =
      ((((long long)b * VSZ + (vx - 1)) * VSZ + (vy - 1)) * VSZ + (vz - 1)) *
      (long long)astride;
  float* q = acc + base;

  __hip_atomic_fetch_add(q + 0, cx,   __ATOMIC_RELAXED, __HIP_MEMORY_SCOPE_AGENT);
  __hip_atomic_fetch_add(q + 1, cy,   __ATOMIC_RELAXED, __HIP_MEMORY_SCOPE_AGENT);
  __hip_atomic_fetch_add(q + 2, cz,   __ATOMIC_RELAXED, __HIP_MEMORY_SCOPE_AGENT);
  __hip_atomic_fetch_add(q + 3, f0,   __ATOMIC_RELAXED, __HIP_MEMORY_SCOPE_AGENT);
  __hip_atomic_fetch_add(q + 4, f1,   __ATOMIC_RELAXED, __HIP_MEMORY_SCOPE_AGENT);
  __hip_atomic_fetch_add(q + 5, f2,   __ATOMIC_RELAXED, __HIP_MEMORY_SCOPE_AGENT);
  __hip_atomic_fetch_add(q + (astride - 1), 1.0f,
                         __ATOMIC_RELAXED, __HIP_MEMORY_SCOPE_AGENT);
}

// ---------------------------------------------------------------------------
// Kernel 3: 256 voxels per block, LDS-staged for fully coalesced I/O.
//   load  : contiguous b128 NT loads of the block's accumulator slab -> LDS
//   math  : per-thread stride-7/10 LDS reads (conflict-free for wave32),
//           true f32 divisions (match reference), grid coords i/100.0f
//   store : 640 x float4 16B-aligned non-temporal b128 stores (full lines)
// In-place (acc==out) is safe: all global loads of the block's region
// complete (dep into ds_store) before the post-barrier global stores.
// ---------------------------------------------------------------------------
__global__ void vg_finalize(const float* __restrict__ acc, int astride,
                            float* __restrict__ out) {
  __shared__ float in_t[2560];              // up to 256 voxels * 10 ch
  __shared__ float out_t[2560];             // 256 voxels * 10 ch
  const int t = threadIdx.x;
  const int vbase = blockIdx.x * 256;       // NVOX % 256 == 0

  // Coalesced load of this block's accumulator slab into LDS.
  const int in4 = 64 * astride;             // float4 per block (448 or 640)
  const v4f* __restrict__ src =
      (const v4f*)(acc + (long long)vbase * astride);
  v4f* in4p = (v4f*)in_t;
  for (int idx = t; idx < in4; idx += 256)
    in4p[idx] = __builtin_nontemporal_load(&src[idx]);
  __syncthreads();

  const int ab = t * astride;
  const float c  = in_t[ab + astride - 1];
  const float mc = fmaxf(c, 1.0f);

  const int v  = vbase + t;
  const int r  = v % 1000000;               // within-batch linear voxel id
  const int i  = r / 10000;
  const int r2 = r % 10000;
  const int j  = r2 / 100;
  const int k  = r2 % 100;

  const int ob = t * 10;
  out_t[ob + 0] = in_t[ab + 0] / mc;
  out_t[ob + 1] = in_t[ab + 1] / mc;
  out_t[ob + 2] = in_t[ab + 2] / mc;
  out_t[ob + 3] = in_t[ab + 3] / mc;
  out_t[ob + 4] = in_t[ab + 4] / mc;
  out_t[ob + 5] = in_t[ab + 5] / mc;
  out_t[ob + 6] = (float)i / 100.0f;
  out_t[ob + 7] = (float)j / 100.0f;
  out_t[ob + 8] = (float)k / 100.0f;
  out_t[ob + 9] = c > 0.0f ? 1.0f : 0.0f;
  __syncthreads();

  // Fully coalesced, 16B-aligned non-temporal stores (2560 dwords/block).
  v4f* __restrict__ dst = (v4f*)(out + (long long)vbase * 10);
  const v4f* o4 = (const v4f*)out_t;
  for (int idx = t; idx < 640; idx += 256)
    __builtin_nontemporal_store(o4[idx], &dst[idx]);
}

// ---------------------------------------------------------------------------
extern "C" void kernel_launch(void* const* d_in, const int* in_sizes, int n_in,
                              void* d_out, int out_size, void* d_ws,
                              size_t ws_size, hipStream_t stream) {
  const float* coords = (const float*)d_in[0];
  const float* feats  = (const float*)d_in[1];
  float* out = (float*)d_out;

  // Prefer a 7-channel accumulator in scratch (less zero+read traffic);
  // otherwise accumulate in-place in d_out using channel 9 as the count.
  const size_t ws_need = (size_t)NVOX * 7 * sizeof(float);
  const bool use_ws = (ws_size >= ws_need) && (d_ws != nullptr);
  float* acc = use_ws ? (float*)d_ws : out;
  const int astride = use_ws ? 7 : 10;

  const long long acc_n4 = ((long long)NVOX * astride) / 4; // 14M or 20M x16B
  const dim3 blk(256);

  vg_zero_async<<<dim3(2048), blk, 0, stream>>>(acc, acc_n4);
  vg_scatter<<<dim3((NPOINTS + 255) / 256), blk, 0, stream>>>(coords, feats,
                                                              acc, astride);
  vg_finalize<<<dim3(NVOX / 256), blk, 0, stream>>>(acc, astride, out);
}